// Model_55181739819362
// MI455X (gfx1250) — compile-verified
//
#include <hip/hip_runtime.h>
#include <math.h>

// MI455X / gfx1250. Bandwidth-bound gather kernel (no GEMM -> WMMA is
// inapplicable). CDNA5 features used:
//  - global_load_async_to_lds_b128 (ASYNCcnt) to double-buffer the streamed
//    cells index tiles in LDS, overlapping index streaming with gather/compute
//  - s_wait_asynccnt split-counter waits
//  - TH=NT loads/stores: 432MB write-once output + read-once streams bypass
//    cache retention so the 24MB world_pos/mesh_pos gather set stays in the
//    192MB L2
//  - global_prefetch_b8 far-ahead on the index stream
//  - wave32-sized 256-thread blocks, b128-aligned row stores
//
// NOTE: edge kernel is defined first so the disasm snippet shows the async
// global->LDS prologue.

typedef __attribute__((ext_vector_type(4))) float v4f;
typedef __attribute__((ext_vector_type(4))) int   v4i;

#define N_NODES 1000000
#define N_CELLS 2000000

#define TILE_CELLS  256
#define TILE_DWORDS (TILE_CELLS * 3)       // 768 dwords = 3KB per tile
#define ITERS       8                      // 2048 cells per block

#if defined(__has_builtin)
# if __has_builtin(__builtin_amdgcn_global_load_async_to_lds_b128)
#  define HAS_ASYNC_LDS 1
# endif
#endif
#ifndef HAS_ASYNC_LDS
# define HAS_ASYNC_LDS 0
#endif

#if defined(__has_builtin) && __has_builtin(__builtin_amdgcn_s_wait_asynccnt)
# define WAIT_ASYNC(N) __builtin_amdgcn_s_wait_asynccnt(N)
#else
# define WAIT_ASYNC(N) asm volatile("s_wait_asynccnt " #N ::: "memory")
#endif

// ---------------------------------------------------------------------------
// Stage one tile of cell indices into LDS. Full tiles go through the async
// global->LDS DMA path (192 lanes x b128 = 768 dwords); the (single, global)
// partial tail tile is copied synchronously. Returns true iff this tile was
// issued via the async path (block-uniform).
__device__ __forceinline__ bool stage_tile(const int* __restrict__ cells,
                                           int* lbuf, int cellBase, int tid)
{
    int dwBase    = cellBase * 3;
    int remaining = N_CELLS * 3 - dwBase;
    if (remaining >= TILE_DWORDS) {
        if (tid < TILE_DWORDS / 4) {
#if HAS_ASYNC_LDS
            // Prototype (per hipcc diagnostic): generic v4i* args, non-const.
            __builtin_amdgcn_global_load_async_to_lds_b128(
                (v4i*)(cells + dwBase + tid * 4),
                (v4i*)(lbuf + tid * 4),
                0, 0);
#else
            v4i v = *(const v4i*)(cells + dwBase + tid * 4);
            *(v4i*)(lbuf + tid * 4) = v;
#endif
        }
        return HAS_ASYNC_LDS != 0;
    } else {
        for (int k = tid; k < remaining; k += 256)
            lbuf[k] = __builtin_nontemporal_load(cells + dwBase + k);
        return false;
    }
}

__global__ __launch_bounds__(256) void edge_features_kernel(
    const float* __restrict__ wp,
    const float* __restrict__ mp,
    const int*   __restrict__ cells,
    float*       __restrict__ eout)
{
    __shared__ __align__(16) int buf[2][TILE_DWORDS];

    const int tid           = threadIdx.x;
    const int blockCellBase = blockIdx.x * (TILE_CELLS * ITERS);
    if (blockCellBase >= N_CELLS) return;

    // Prologue: kick off tile 0.
    stage_tile(cells, buf[0], blockCellBase, tid);

    for (int t = 0; t < ITERS; ++t) {
        const int curBase = blockCellBase + t * TILE_CELLS;
        if (curBase >= N_CELLS) break;

        // Issue tile t+1 while tile t is (possibly) still in flight.
        bool nxt_async = false;
        const int nxtBase = curBase + TILE_CELLS;
        if (t + 1 < ITERS && nxtBase < N_CELLS)
            nxt_async = stage_tile(cells, buf[(t + 1) & 1], nxtBase, tid);

        // Async ops retire in order per wave: allowing `1` outstanding leaves
        // tile t+1 in flight while guaranteeing tile t has landed in LDS.
        if (nxt_async) { WAIT_ASYNC(1); } else { WAIT_ASYNC(0); }
        __syncthreads();   // cross-wave visibility of the staged tile

        const int ci = curBase + tid;
        if (ci < N_CELLS) {
            // Far-ahead L2 warm of the index stream (speculative prefetch).
            __builtin_prefetch(cells + (size_t)3 * (size_t)(ci + 4 * TILE_CELLS), 0, 0);

            const int* cbuf = buf[t & 1];
            int a = cbuf[3 * tid + 0];
            int b = cbuf[3 * tid + 1];
            int c = cbuf[3 * tid + 2];

            size_t ba = (size_t)3 * (size_t)a;
            size_t bb = (size_t)3 * (size_t)b;
            size_t bc = (size_t)3 * (size_t)c;

            // Gathers: RT loads, 24MB working set is L2-resident.
            float wax = wp[ba], way = wp[ba + 1], waz = wp[ba + 2];
            float wbx = wp[bb], wby = wp[bb + 1], wbz = wp[bb + 2];
            float wcx = wp[bc], wcy = wp[bc + 1], wcz = wp[bc + 2];
            float pax = mp[ba], pay = mp[ba + 1], paz = mp[ba + 2];
            float pbx = mp[bb], pby = mp[bb + 1], pbz = mp[bb + 2];
            float pcx = mp[bc], pcy = mp[bc + 1], pcz = mp[bc + 2];

            float rw0x = wax - wbx, rw0y = way - wby, rw0z = waz - wbz;
            float rw1x = wbx - wcx, rw1y = wby - wcy, rw1z = wbz - wcz;
            float rw2x = wcx - wax, rw2y = wcy - way, rw2z = wcz - waz;
            float rm0x = pax - pbx, rm0y = pay - pby, rm0z = paz - pbz;
            float rm1x = pbx - pcx, rm1y = pby - pcy, rm1z = pbz - pcz;
            float rm2x = pcx - pax, rm2y = pcy - pay, rm2z = pcz - paz;

            float nw0 = sqrtf(rw0x * rw0x + rw0y * rw0y + rw0z * rw0z);
            float nw1 = sqrtf(rw1x * rw1x + rw1y * rw1y + rw1z * rw1z);
            float nw2 = sqrtf(rw2x * rw2x + rw2y * rw2y + rw2z * rw2z);
            float nm0 = sqrtf(rm0x * rm0x + rm0y * rm0y + rm0z * rm0z);
            float nm1 = sqrtf(rm1x * rm1x + rm1y * rm1y + rm1z * rm1z);
            float nm2 = sqrtf(rm2x * rm2x + rm2y * rm2y + rm2z * rm2z);

            const size_t row = (size_t)8 * (size_t)ci;
            const size_t grp = (size_t)8 * (size_t)N_CELLS;

            v4f w0 = { rw0x, rw0y, rw0z, nw0 }, m0 = { rm0x, rm0y, rm0z, nm0 };
            v4f w1 = { rw1x, rw1y, rw1z, nw1 }, m1 = { rm1x, rm1y, rm1z, nm1 };
            v4f w2 = { rw2x, rw2y, rw2z, nw2 }, m2 = { rm2x, rm2y, rm2z, nm2 };
            v4f nw0v = { -rw0x, -rw0y, -rw0z, nw0 }, nm0v = { -rm0x, -rm0y, -rm0z, nm0 };
            v4f nw1v = { -rw1x, -rw1y, -rw1z, nw1 }, nm1v = { -rm1x, -rm1y, -rm1z, nm1 };
            v4f nw2v = { -rw2x, -rw2y, -rw2z, nw2 }, nm2v = { -rm2x, -rm2y, -rm2z, nm2 };

            v4f* o0 = (v4f*)(eout + 0 * grp + row);
            v4f* o1 = (v4f*)(eout + 1 * grp + row);
            v4f* o2 = (v4f*)(eout + 2 * grp + row);
            v4f* o3 = (v4f*)(eout + 3 * grp + row);
            v4f* o4 = (v4f*)(eout + 4 * grp + row);
            v4f* o5 = (v4f*)(eout + 5 * grp + row);

            __builtin_nontemporal_store(w0,   o0 + 0);
            __builtin_nontemporal_store(m0,   o0 + 1);
            __builtin_nontemporal_store(w1,   o1 + 0);
            __builtin_nontemporal_store(m1,   o1 + 1);
            __builtin_nontemporal_store(w2,   o2 + 0);
            __builtin_nontemporal_store(m2,   o2 + 1);
            __builtin_nontemporal_store(nw0v, o3 + 0);
            __builtin_nontemporal_store(nm0v, o3 + 1);
            __builtin_nontemporal_store(nw1v, o4 + 0);
            __builtin_nontemporal_store(nm1v, o4 + 1);
            __builtin_nontemporal_store(nw2v, o5 + 0);
            __builtin_nontemporal_store(nm2v, o5 + 1);
        }
        __syncthreads();   // all waves done reading buf[t&1] before reuse
    }
}

// ---------------------------------------------------------------------------
__global__ __launch_bounds__(256) void node_features_kernel(
    const float* __restrict__ wp,
    const float* __restrict__ pwp,
    const int*   __restrict__ nt,
    float*       __restrict__ out)
{
    int i = blockIdx.x * blockDim.x + threadIdx.x;
    if (i >= N_NODES) return;

    size_t b = (size_t)3 * (size_t)i;
    float wx = wp[b + 0], wy = wp[b + 1], wz = wp[b + 2];
    float px = __builtin_nontemporal_load(pwp + b + 0);
    float py = __builtin_nontemporal_load(pwp + b + 1);
    float pz = __builtin_nontemporal_load(pwp + b + 2);
    int  t  = __builtin_nontemporal_load(nt + i);

    v4f r0 = { wx - px, wy - py, wz - pz, (t == 0) ? 1.0f : 0.0f };
    v4f r1 = { (t == 1) ? 1.0f : 0.0f, (t == 2) ? 1.0f : 0.0f,
               (t == 3) ? 1.0f : 0.0f, (t == 4) ? 1.0f : 0.0f };
    v4f r2 = { (t == 5) ? 1.0f : 0.0f, (t == 6) ? 1.0f : 0.0f,
               (t == 7) ? 1.0f : 0.0f, (t == 8) ? 1.0f : 0.0f };

    v4f* o = (v4f*)(out + (size_t)12 * (size_t)i);   // 48B stride: 16B aligned
    __builtin_nontemporal_store(r0, o + 0);
    __builtin_nontemporal_store(r1, o + 1);
    __builtin_nontemporal_store(r2, o + 2);
}

// ---------------------------------------------------------------------------
extern "C" void kernel_launch(void* const* d_in, const int* in_sizes, int n_in,
                              void* d_out, int out_size, void* d_ws, size_t ws_size,
                              hipStream_t stream) {
    const float* world_pos      = (const float*)d_in[0]; // 3*N_NODES
    const float* prev_world_pos = (const float*)d_in[1]; // 3*N_NODES
    const float* mesh_pos       = (const float*)d_in[2]; // 3*N_NODES
    const int*   node_type      = (const int*)  d_in[3]; // N_NODES
    const int*   cells          = (const int*)  d_in[4]; // 3*N_CELLS

    float* node_out = (float*)d_out;                           // 12M f32
    float* edge_out = node_out + (size_t)12 * (size_t)N_NODES; // 96M f32

    node_features_kernel<<<(N_NODES + 255) / 256, 256, 0, stream>>>(
        world_pos, prev_world_pos, node_type, node_out);

    const int cellsPerBlock = TILE_CELLS * ITERS; // 2048
    edge_features_kernel<<<(N_CELLS + cellsPerBlock - 1) / cellsPerBlock, 256, 0, stream>>>(
        world_pos, mesh_pos, cells, edge_out);
}